// SSIMLoss_37297495998709
// MI455X (gfx1250) — compile-verified
//
#include <hip/hip_runtime.h>
#include <hip/hip_bf16.h>

// ---------------------------------------------------------------------------
// Fused SSIM loss for (16,3,512,512) f32 images on gfx1250 (MI455X).
// Memory-bound (~100MB in, scalar out): single pass over inputs.
//  - async global->LDS staging (global_load_async_to_lds_b128, ASYNCcnt)
//  - vertical 11-tap box pass as chained V_WMMA_F32_16X16X4_F32
//  - horizontal 11-tap pass as rolling sums + SSIM map + reduction
// ---------------------------------------------------------------------------

typedef __attribute__((ext_vector_type(2))) float v2f;
typedef __attribute__((ext_vector_type(8))) float v8f;

#define IMG_H 512
#define IMG_W 512
#define TILE  64
#define IN_W  80          // staged cols: [c0-8, c0+72), 16B aligned
#define IN_H  76          // 74 valid rows + 2 zero rows (K=28 padding)
#define LXW   80          // lx/ly row stride (floats) -> 320B, 16B aligned
#define VSW   84          // vsum row stride (floats), de-conflicted banks
#define VS_Q  (64 * VSW)  // per-quantity vsum plane (floats)
#define LX_OFF 0
#define LY_OFF (IN_H * LXW)            // 6080
#define VS_OFF (2 * IN_H * LXW)        // 12160
#define SMEM_FLOATS (VS_OFF + 5 * VS_Q) // 39040 -> 156,160 bytes
#define N_TOTAL 12582912.0             // 16*3*512*512

__device__ __forceinline__ void async_ld128(unsigned lds_off, const float* base,
                                            unsigned byte_off) {
  // vdst = LDS byte address, vaddr = 32-bit byte offset, saddr = 64-bit base
  asm volatile("global_load_async_to_lds_b128 %0, %1, %2"
               :: "v"(lds_off), "v"(byte_off), "s"(base)
               : "memory");
}

__global__ void __launch_bounds__(256)
ssim_tile_kernel(const float* __restrict__ img1, const float* __restrict__ img2,
                 double* __restrict__ acc) {
  extern __shared__ float smem[];
  float* lx = smem + LX_OFF;
  float* ly = smem + LY_OFF;
  float* vs = smem + VS_OFF;

  const int t   = threadIdx.x;
  const int c0  = blockIdx.x * TILE;
  const int r0  = blockIdx.y * TILE;
  const unsigned pbase = blockIdx.z * (unsigned)(IMG_H * IMG_W);

  // ---- Stage 0: zero-fill halos that async loads will not cover ----------
  const bool border = (c0 == 0) | (r0 == 0) |
                      (c0 + TILE == IMG_W) | (r0 + TILE == IMG_H);
  if (border) {
    float4* p = (float4*)smem;                 // lx+ly = 12160 floats
    for (int i = t; i < (2 * IN_H * LXW) / 4; i += 256)
      p[i] = make_float4(0.f, 0.f, 0.f, 0.f);
  } else {
    // only the two K-padding rows (74,75) of each tile need zeroing
    for (int i = t; i < 2 * 2 * LXW; i += 256) {
      if (i < 2 * LXW) lx[74 * LXW + i] = 0.f;
      else             ly[74 * LXW + (i - 2 * LXW)] = 0.f;
    }
  }
  __syncthreads();   // all zero stores complete before async engine writes

  // ---- Stage A: async global -> LDS (74 rows x 20 float4 chunks) ---------
  {
    const int gr0 = r0 - 5;
    const int gc0 = c0 - 8;                    // multiple of 4, 16B aligned
    for (int idx = t; idx < 74 * 20; idx += 256) {
      const int row = idx / 20;
      const int chk = idx - row * 20;
      const int gr = gr0 + row;
      const int gc = gc0 + chk * 4;
      // chunk is fully in-bounds or fully out (gc % 4 == 0, borders at 0/512)
      if ((unsigned)gr < (unsigned)IMG_H && (unsigned)gc <= (unsigned)(IMG_W - 4)) {
        const unsigned off = (pbase + (unsigned)gr * IMG_W + (unsigned)gc) * 4u;
        const unsigned lofs = (unsigned)(row * LXW + chk * 4) * 4u;
        async_ld128((unsigned)(uintptr_t)lx + lofs, img1, off);
        async_ld128((unsigned)(uintptr_t)ly + lofs, img2, off);
      }
    }
  }
  asm volatile("s_wait_asynccnt 0" ::: "memory");
  __syncthreads();

  // ---- Stage B: vertical 11-tap box sums via WMMA f32 16x16x4 ------------
  // vsum[r][c] = (1/11) * sum_{k=0..10} in[r+k][c]  as  D = A(16x28) x B(28x16)
  const int wave = t >> 5;
  const int lane = t & 31;
  const int lrow = lane & 15;   // A: M row / B,D: N column
  const int lhi  = lane >> 4;   // lane half selects K (A,B) / M+8 (D)

  for (int pos = wave; pos < 20; pos += 8) {     // 4 row-tiles x 5 col-tiles
    const int r0t = (pos / 5) * 16;
    const int cb  = (pos % 5) * 16;
    v8f accv[5] = {};
    for (int s = 0; s < 7; ++s) {                // K = 28 in steps of 4
      const int j = 4 * s + lhi * 2;             // global K of this lane half
      const float* px = lx + (r0t + j) * LXW + cb + lrow;
      const float* py = ly + (r0t + j) * LXW + cb + lrow;
      const float x0 = px[0],   x1 = px[LXW];
      const float y0 = py[0],   y1 = py[LXW];
      const float w = 1.0f / 11.0f;
      const float a0 = (j     >= lrow && j     <= lrow + 10) ? w : 0.0f;
      const float a1 = (j + 1 >= lrow && j + 1 <= lrow + 10) ? w : 0.0f;
      const v2f A = {a0, a1};
      v2f B;
      B = (v2f){x0, x1};
      accv[0] = __builtin_amdgcn_wmma_f32_16x16x4_f32(false, A, false, B,
                                                      (short)0, accv[0], false, false);
      B = (v2f){y0, y1};
      accv[1] = __builtin_amdgcn_wmma_f32_16x16x4_f32(false, A, false, B,
                                                      (short)0, accv[1], false, false);
      B = (v2f){x0 * x0, x1 * x1};
      accv[2] = __builtin_amdgcn_wmma_f32_16x16x4_f32(false, A, false, B,
                                                      (short)0, accv[2], false, false);
      B = (v2f){y0 * y0, y1 * y1};
      accv[3] = __builtin_amdgcn_wmma_f32_16x16x4_f32(false, A, false, B,
                                                      (short)0, accv[3], false, false);
      B = (v2f){x0 * y0, x1 * y1};
      accv[4] = __builtin_amdgcn_wmma_f32_16x16x4_f32(false, A, false, B,
                                                      (short)0, accv[4], false, false);
    }
    // D layout: VGPR v, lanes 0-15 -> (M=v, N=lane); lanes 16-31 -> (M=v+8)
    float* vp = vs + (r0t + lhi * 8) * VSW + cb + lrow;
    for (int q = 0; q < 5; ++q)
      for (int v = 0; v < 8; ++v)
        vp[q * VS_Q + v * VSW] = accv[q][v];
  }
  __syncthreads();

  // ---- Stage C: horizontal 11-tap rolling sums + SSIM + reduction --------
  const int r  = t >> 2;           // output row 0..63
  const int cg = (t & 3) * 16;     // 16-column group
  const float inv11 = 1.0f / 11.0f;
  const float C1v = 0.01f * 0.01f, C2v = 0.03f * 0.03f;

  const float* vr = vs + r * VSW + cg;
  float s0 = 0.f, s1 = 0.f, s2 = 0.f, s3 = 0.f, s4 = 0.f;
#pragma unroll
  for (int k = 3; k <= 13; ++k) {
    s0 += vr[k];
    s1 += vr[VS_Q + k];
    s2 += vr[2 * VS_Q + k];
    s3 += vr[3 * VS_Q + k];
    s4 += vr[4 * VS_Q + k];
  }
  float lsum = 0.f;
#pragma unroll
  for (int c = 0; c < 16; ++c) {
    const float mu1 = s0 * inv11, mu2 = s1 * inv11;
    const float exx = s2 * inv11, eyy = s3 * inv11, exy = s4 * inv11;
    const float mu1sq = mu1 * mu1, mu2sq = mu2 * mu2, mu12 = mu1 * mu2;
    const float sg1 = exx - mu1sq, sg2 = eyy - mu2sq, sg12 = exy - mu12;
    const float num = (2.f * mu12 + C1v) * (2.f * sg12 + C2v);
    const float den = (mu1sq + mu2sq + C1v) * (sg1 + sg2 + C2v);
    lsum += num / den;
    if (c < 15) {
      const int ad = c + 14, sb = c + 3;
      s0 += vr[ad] - vr[sb];
      s1 += vr[VS_Q + ad] - vr[VS_Q + sb];
      s2 += vr[2 * VS_Q + ad] - vr[2 * VS_Q + sb];
      s3 += vr[3 * VS_Q + ad] - vr[3 * VS_Q + sb];
      s4 += vr[4 * VS_Q + ad] - vr[4 * VS_Q + sb];
    }
  }
  // wave32 tree reduction, then one f64 atomic per wave
  for (int off = 16; off > 0; off >>= 1)
    lsum += __shfl_down(lsum, off, 32);
  if (lane == 0)
    atomicAdd(acc, (double)lsum);
}

__global__ void ssim_zero_kernel(double* acc) { acc[0] = 0.0; }

__global__ void ssim_final_kernel(const double* acc, float* out) {
  out[0] = 1.0f - (float)(acc[0] / N_TOTAL);
}

extern "C" void kernel_launch(void* const* d_in, const int* in_sizes, int n_in,
                              void* d_out, int out_size, void* d_ws, size_t ws_size,
                              hipStream_t stream) {
  const float* img1 = (const float*)d_in[0];
  const float* img2 = (const float*)d_in[1];
  float*  out = (float*)d_out;
  double* acc = (double*)d_ws;

  ssim_zero_kernel<<<1, 1, 0, stream>>>(acc);
  dim3 grid(IMG_W / TILE, IMG_H / TILE, 48);   // 8 x 8 x (16*3)
  ssim_tile_kernel<<<grid, 256, SMEM_FLOATS * 4, stream>>>(img1, img2, acc);
  ssim_final_kernel<<<1, 1, 0, stream>>>(acc, out);
}